// GNN_64501818851480
// MI455X (gfx1250) — compile-verified
//
#include <hip/hip_runtime.h>
#include <hip/hip_bf16.h>

#define NN   50000
#define EE   800000
#define DD   64
#define ETOT (EE + NN)
#define MT   (NN / 16)   /* 3125 row panels, exact */

typedef __attribute__((ext_vector_type(2))) float v2f;
typedef __attribute__((ext_vector_type(8))) float v8f;

// ---------------- utility ----------------
__global__ void zero_i32(int* __restrict__ p, int n) {
  int i = blockIdx.x * blockDim.x + threadIdx.x;
  if (i < n) p[i] = 0;
}

// ---------------- CSR build ----------------
__global__ void count_edges(const int* __restrict__ ei, int* __restrict__ cnt) {
  int e = blockIdx.x * blockDim.x + threadIdx.x;
  if (e >= ETOT) return;
  int s = (e < EE) ? ei[e] : (e - EE);        // self-loop tail: src = n
  atomicAdd(&cnt[s], 1);
}

// single-block exclusive scan over NN counters (1024 threads, chunked)
__global__ void scan_block(const int* __restrict__ cnt, int* __restrict__ start) {
  __shared__ int buf[1024];
  __shared__ int carry_s;
  if (threadIdx.x == 0) carry_s = 0;
  __syncthreads();
  for (int base = 0; base < NN; base += 1024) {
    int i = base + threadIdx.x;
    int v = (i < NN) ? cnt[i] : 0;
    buf[threadIdx.x] = v;
    __syncthreads();
    for (int off = 1; off < 1024; off <<= 1) {
      int t = (threadIdx.x >= off) ? buf[threadIdx.x - off] : 0;
      __syncthreads();
      buf[threadIdx.x] += t;
      __syncthreads();
    }
    int inc   = buf[threadIdx.x];
    int carry = carry_s;
    if (i < NN) start[i] = carry + inc - v;   // exclusive
    int total = buf[1023];
    __syncthreads();
    if (threadIdx.x == 0) carry_s = carry + total;
    __syncthreads();
  }
  if (threadIdx.x == 0) start[NN] = carry_s;
}

__global__ void fill_edges(const int* __restrict__ ei, const int* __restrict__ start,
                           int* __restrict__ cur, int* __restrict__ edst) {
  int e = blockIdx.x * blockDim.x + threadIdx.x;
  if (e >= ETOT) return;
  int s, d;
  if (e < EE) { s = ei[e]; d = ei[EE + e]; }
  else        { s = e - EE; d = e - EE; }     // self loop
  int pos = start[s] + atomicAdd(&cur[s], 1);
  edst[pos] = d;
}

// ---------------- embedding lookup ----------------
__global__ void embed(const int* __restrict__ x, const float* __restrict__ emb,
                      float* __restrict__ f) {
  int i = blockIdx.x * blockDim.x + threadIdx.x;
  if (i >= NN * DD) return;
  f[i] = emb[x[i >> 6] * DD + (i & 63)];
}

// ---- fused GEMM + attention score ----
// one wave per 16-row panel: g[16,128] = f[16,64] @ W[64,128] + b  (fp32 WMMA)
// epilogue: ls[n,h] = leaky_relu(g[n,h,:].att[h]) computed from the registers.
__global__ void gemm_wmma(const float* __restrict__ F, const float* __restrict__ W,
                          const float* __restrict__ b, const float* __restrict__ attf,
                          float* __restrict__ G, float* __restrict__ ls) {
  int t    = blockIdx.x * blockDim.x + threadIdx.x;
  int wave = t >> 5;
  int lane = t & 31;
  if (wave >= MT) return;                 // wave-uniform guard (tail block)
  int row0 = wave * 16;
  int ml = lane & 15;                     // M (for A) / N (for B/C/D)
  int kh = (lane >> 4) * 2;               // lanes 16-31 hold K+2 halves

  v8f acc[8] = {{}, {}, {}, {}, {}, {}, {}, {}};
  const float* arow = F + (size_t)(row0 + ml) * 64;
#pragma unroll 4
  for (int k0 = 0; k0 < 64; k0 += 4) {
    float2 av = *(const float2*)(arow + k0 + kh);   // A: (M=ml, K=k0+kh, +1)
    v2f a; a.x = av.x; a.y = av.y;
    const float* wr = W + (size_t)(k0 + kh) * 128 + ml;
#pragma unroll
    for (int j = 0; j < 8; ++j) {                   // 8 column tiles, A reused
      v2f bb;
      bb.x = wr[j * 16];                            // B: (K=k0+kh,   N=j*16+ml)
      bb.y = wr[128 + j * 16];                      // B: (K=k0+kh+1, N=j*16+ml)
      acc[j] = __builtin_amdgcn_wmma_f32_16x16x4_f32(false, a, false, bb,
                                                     (short)0, acc[j], false, false);
    }
  }

  // bias into accumulators first (scores must see fW + b)
#pragma unroll
  for (int j = 0; j < 8; ++j) {
    float bias = b[j * 16 + ml];
#pragma unroll
    for (int v = 0; v < 8; ++v) acc[j][v] += bias;
  }

  // store 16x128 panel
  int mb = (lane >> 4) * 8;                          // lanes>=16 hold M=8..15
#pragma unroll
  for (int j = 0; j < 8; ++j)
#pragma unroll
    for (int v = 0; v < 8; ++v)
      G[(size_t)(row0 + mb + v) * 128 + j * 16 + ml] = acc[j][v];

  // fused attention scores: per row, dot over cols (h0 = tiles 0-3, h1 = 4-7)
  float p0[8], p1[8];
#pragma unroll
  for (int v = 0; v < 8; ++v) { p0[v] = 0.f; p1[v] = 0.f; }
#pragma unroll
  for (int j = 0; j < 8; ++j) {
    float av = attf[j * 16 + ml];                    // att flat matches col layout
#pragma unroll
    for (int v = 0; v < 8; ++v) {
      if (j < 4) p0[v] += acc[j][v] * av;
      else       p1[v] += acc[j][v] * av;
    }
  }
#pragma unroll
  for (int v = 0; v < 8; ++v) {
#pragma unroll
    for (int off = 1; off < 16; off <<= 1) {         // reduce within 16-lane half
      p0[v] += __shfl_xor(p0[v], off, 32);
      p1[v] += __shfl_xor(p1[v], off, 32);
    }
  }
  if ((lane & 15) == 0) {                            // lane 0 -> rows 0-7, lane 16 -> rows 8-15
    int rbase = row0 + mb;
#pragma unroll
    for (int v = 0; v < 8; ++v) {
      float s0 = p0[v], s1 = p1[v];
      s0 = (s0 > 0.f) ? s0 : 0.2f * s0;              // leaky_relu slope 0.2
      s1 = (s1 > 0.f) ? s1 : 0.2f * s1;
      *(float2*)(ls + (size_t)(rbase + v) * 2) = make_float2(s0, s1);
    }
  }
}

// ---------------- wave-per-node softmax + weighted aggregation ----------------
__global__ void aggregate(const float* __restrict__ G, const float* __restrict__ ls,
                          const int* __restrict__ start, const int* __restrict__ edst,
                          const float* __restrict__ bias, float* __restrict__ out,
                          int do_relu) {
  int t = blockIdx.x * blockDim.x + threadIdx.x;
  int n = t >> 5, lane = t & 31;
  if (n >= NN) return;
  int s0 = start[n], s1 = start[n + 1];

  // pass 1: segment max per head
  float m0 = -1e30f, m1 = -1e30f;
  for (int j = s0 + lane; j < s1; j += 32) {
    float2 a = *(const float2*)(ls + (size_t)edst[j] * 2);
    m0 = fmaxf(m0, a.x); m1 = fmaxf(m1, a.y);
  }
#pragma unroll
  for (int off = 16; off >= 1; off >>= 1) {
    m0 = fmaxf(m0, __shfl_xor(m0, off, 32));
    m1 = fmaxf(m1, __shfl_xor(m1, off, 32));
  }
  // pass 2: segment sum of exp
  float d0 = 0.f, d1 = 0.f;
  for (int j = s0 + lane; j < s1; j += 32) {
    float2 a = *(const float2*)(ls + (size_t)edst[j] * 2);
    d0 += __expf(a.x - m0); d1 += __expf(a.y - m1);
  }
#pragma unroll
  for (int off = 16; off >= 1; off >>= 1) {
    d0 += __shfl_xor(d0, off, 32);
    d1 += __shfl_xor(d1, off, 32);
  }
  d0 += 1e-16f; d1 += 1e-16f;

  // pass 3: whole wave walks the adjacency; lane owns 4 columns (lanes<16: h0, >=16: h1)
  float myM  = (lane < 16) ? m0 : m1;
  float invD = 1.f / ((lane < 16) ? d0 : d1);
  int   cb   = lane * 4;
  float a0 = 0.f, a1 = 0.f, a2 = 0.f, a3 = 0.f;
  for (int j = s0; j < s1; ++j) {
    int dn = edst[j];
    float2 av = *(const float2*)(ls + (size_t)dn * 2);
    float aa = (lane < 16) ? av.x : av.y;
    float w  = __expf(aa - myM) * invD;
    float4 gv = *(const float4*)(G + (size_t)dn * 128 + cb);   // coalesced 512B row
    a0 += w * gv.x; a1 += w * gv.y; a2 += w * gv.z; a3 += w * gv.w;
  }
  // sum of x_i over segment = deg * g[n]
  float degf = (float)(s1 - s0);
  float4 gn = *(const float4*)(G + (size_t)n * 128 + cb);
  a0 += degf * gn.x; a1 += degf * gn.y; a2 += degf * gn.z; a3 += degf * gn.w;

  // mean over heads (pairs lane L with lane L+16), + bias, optional relu
  float o0 = 0.5f * (a0 + __shfl_xor(a0, 16, 32));
  float o1 = 0.5f * (a1 + __shfl_xor(a1, 16, 32));
  float o2 = 0.5f * (a2 + __shfl_xor(a2, 16, 32));
  float o3 = 0.5f * (a3 + __shfl_xor(a3, 16, 32));
  if (lane < 16) {
    float4 bv = *(const float4*)(bias + cb);
    o0 += bv.x; o1 += bv.y; o2 += bv.z; o3 += bv.w;
    if (do_relu) {
      o0 = fmaxf(o0, 0.f); o1 = fmaxf(o1, 0.f);
      o2 = fmaxf(o2, 0.f); o3 = fmaxf(o3, 0.f);
    }
    *(float4*)(out + (size_t)n * 64 + cb) = make_float4(o0, o1, o2, o3);
  }
}

// ---------------- host ----------------
extern "C" void kernel_launch(void* const* d_in, const int* in_sizes, int n_in,
                              void* d_out, int out_size, void* d_ws, size_t ws_size,
                              hipStream_t stream) {
  const int*   x      = (const int*)d_in[0];
  const int*   ei     = (const int*)d_in[1];   // [2,E]: row0=src, row1=dst
  const float* emb    = (const float*)d_in[2]; // [2,64]
  const float* Ws     = (const float*)d_in[3]; // [2,64,128]
  const float* bs     = (const float*)d_in[4]; // [2,128]
  const float* atts   = (const float*)d_in[5]; // [2,2,64]
  const float* biases = (const float*)d_in[6]; // [2,64]
  float* outp = (float*)d_out;

  char* ws = (char*)d_ws;
  size_t off = 0;
  auto alloc = [&](size_t bytes) -> void* {
    void* p = ws + off;
    off = (off + bytes + 255) & ~(size_t)255;
    return p;
  };
  int*   cnt   = (int*)alloc((size_t)NN * 4);
  int*   cur   = (int*)alloc((size_t)NN * 4);
  int*   start = (int*)alloc((size_t)(NN + 1) * 4);
  int*   edst  = (int*)alloc((size_t)ETOT * 4);
  float* f0    = (float*)alloc((size_t)NN * 64 * 4);
  float* f1    = (float*)alloc((size_t)NN * 64 * 4);
  float* g     = (float*)alloc((size_t)NN * 128 * 4);
  float* ls    = (float*)alloc((size_t)NN * 2 * 4);

  const int B = 256;
  // CSR build
  zero_i32<<<(NN + B - 1) / B, B, 0, stream>>>(cnt, NN);
  count_edges<<<(ETOT + B - 1) / B, B, 0, stream>>>(ei, cnt);
  scan_block<<<1, 1024, 0, stream>>>(cnt, start);
  zero_i32<<<(NN + B - 1) / B, B, 0, stream>>>(cur, NN);
  fill_edges<<<(ETOT + B - 1) / B, B, 0, stream>>>(ei, start, cur, edst);

  // input embeddings
  embed<<<(NN * DD + B - 1) / B, B, 0, stream>>>(x, emb, f0);

  const int gemm_blocks = (MT * 32 + B - 1) / B;   // 3125 waves, 8 per block
  const int node_blocks = (NN * 32) / B;           // 6250, exact

  // layer 0: GEMM+scores fused, then aggregate (with relu)
  gemm_wmma<<<gemm_blocks, B, 0, stream>>>(f0, Ws, bs, atts, g, ls);
  aggregate<<<node_blocks, B, 0, stream>>>(g, ls, start, edst, biases, f1, 1);

  // layer 1: writes final output, no relu (JK='last')
  gemm_wmma<<<gemm_blocks, B, 0, stream>>>(f1, Ws + 64 * 128, bs + 128,
                                           atts + 128, g, ls);
  aggregate<<<node_blocks, B, 0, stream>>>(g, ls, start, edst, biases + 64, outp, 0);
}